// PreciseBetaCDF_31129922961831
// MI455X (gfx1250) — compile-verified
//
#include <hip/hip_runtime.h>
#include <hip/hip_bf16.h>

// PreciseBetaCDF on gfx1250 (MI455X).
//
// Compute-bound elementwise continued fraction (no matmul structure -> WMMA
// inapplicable). CDNA5-targeted structure:
//  * Fused Mobius pair-step: ONE v_rcp_f32 per (even,odd) CF pair, shared by
//    both the d-update and the f-multiplier; rcp runs on the TRANS pipe and
//    co-executes with VALU -> loop is pure-VALU bound (~18 ops/pair/elem).
//  * 4 independent CF chains per lane (float4) for latency hiding.
//  * gfx1250 async global->LDS staging (global_load_async_to_lds_b128 +
//    s_wait_asynccnt) for the inputs, read back via ds_load_b128.
//  * wave32-sized blocks (256 threads = 8 waves per WGP-pair).

#define TPB 256
#define CF_PAIRS 500   // 1000 reference iterations = 500 (even,odd) pairs

__global__ __launch_bounds__(TPB) void
PreciseBetaCDF_kernel(const float* __restrict__ gz,
                      const float* __restrict__ ga,
                      const float* __restrict__ gb,
                      float* __restrict__ gout,
                      int n)
{
    __shared__ float4 smem[3 * TPB];   // z | alpha | beta tiles (12 KB)

    const int tid = threadIdx.x;
    const long long i4   = (long long)blockIdx.x * TPB + tid;
    const long long base = i4 * 4;
    if (base >= (long long)n) return;

    float x[4], a[4], b[4];
    const bool vec = (base + 3) < (long long)n;

    if (vec) {
        // --- CDNA5 async global->LDS staging (ASYNCcnt-tracked) ---
        unsigned long long az = (unsigned long long)(gz + base);
        unsigned long long aa = (unsigned long long)(ga + base);
        unsigned long long ab = (unsigned long long)(gb + base);
        // Flat LDS addresses truncate to the LDS byte offset (ISA 10.2).
        unsigned oz = (unsigned)(uintptr_t)&smem[tid];
        unsigned oa = (unsigned)(uintptr_t)&smem[TPB + tid];
        unsigned ob = (unsigned)(uintptr_t)&smem[2 * TPB + tid];
        asm volatile("global_load_async_to_lds_b128 %0, %1, off"
                     :: "v"(oz), "v"(az) : "memory");
        asm volatile("global_load_async_to_lds_b128 %0, %1, off"
                     :: "v"(oa), "v"(aa) : "memory");
        asm volatile("global_load_async_to_lds_b128 %0, %1, off"
                     :: "v"(ob), "v"(ab) : "memory");
        asm volatile("s_wait_asynccnt 0" ::: "memory");

        float4 zv = smem[tid];
        float4 av = smem[TPB + tid];
        float4 bv = smem[2 * TPB + tid];
        x[0] = zv.x; x[1] = zv.y; x[2] = zv.z; x[3] = zv.w;
        a[0] = av.x; a[1] = av.y; a[2] = av.z; a[3] = av.w;
        b[0] = bv.x; b[1] = bv.y; b[2] = bv.z; b[3] = bv.w;
    } else {
        // tail (never taken for 2048x2048, kept for generality)
        #pragma unroll
        for (int k = 0; k < 4; ++k) {
            long long idx = base + k;
            bool ok = idx < (long long)n;
            x[k] = ok ? gz[idx] : 0.5f;
            a[k] = ok ? ga[idx] : 1.0f;
            b[k] = ok ? gb[idx] : 1.0f;
        }
    }

    // ---- continued fraction: 4 independent chains, fused pair steps ----
    // Incremental per-chain state: am = a+m, apbm = (a+b)+m, bmm = b-m,
    // s = a+2m.  One rcp per pair:
    //   t  = Qe + Pe*d            (Pe = -(am*apbm)*x, Qe = s(s+1))
    //   u  = Qo*t + Po*Qe         (Po = (m*bmm)*x,    Qo = (s-1)s)
    //   d' = Qo*t/u,   f *= Qe*Qo/u     (shared rcp(u))
    float apb[4], d[4], f[4], am[4], apbm[4], bmm[4], s[4];
    #pragma unroll
    for (int k = 0; k < 4; ++k) {
        x[k]    = fminf(fmaxf(x[k], 0.0f), 1.0f);      // clip(z,0,1)
        apb[k]  = a[k] + b[k];
        float ap1 = a[k] + 1.0f;
        // d0 = 1/(1-(a+b)x/(a+1)) = (a+1)/((a+1)-(a+b)x)
        d[k]  = ap1 * __builtin_amdgcn_rcpf(fmaf(-apb[k], x[k], ap1));
        f[k]  = 0.0f;   // faithful to reference's f0 = 0 (=> f stays 0/NaN)
        am[k]   = a[k];
        apbm[k] = apb[k];
        bmm[k]  = b[k];
        s[k]    = a[k];
    }

    float m = 0.0f;
    #pragma unroll 1
    for (int it = 0; it < CF_PAIRS; ++it) {
        #pragma unroll
        for (int k = 0; k < 4; ++k) {
            float Qe = fmaf(s[k], s[k],  s[k]);        // s(s+1)
            float Qo = fmaf(s[k], s[k], -s[k]);        // (s-1)s
            float Pe = (am[k] * apbm[k]) * x[k];       // negated via fma below
            float Po = (m * bmm[k]) * x[k];
            float t  = fmaf(-Pe, d[k], Qe);            // Qe + (-Pe)*d ... Pe carries sign
            float u  = fmaf(Qo, t, Po * Qe);
            float r  = __builtin_amdgcn_rcpf(u);       // one TRANS op per pair
            d[k]     = (Qo * t) * r;
            f[k]    *= (Qe * Qo) * r;                  // f *= d1*d2
            am[k]   += 1.0f;
            apbm[k] += 1.0f;
            bmm[k]  -= 1.0f;
            s[k]    += 2.0f;
        }
        m += 1.0f;
    }

    // ---- epilogue: beta_x * f / a, edge handling ----
    float r[4];
    #pragma unroll
    for (int k = 0; k < 4; ++k) {
        float lb = lgammaf(apb[k]) - lgammaf(a[k]) - lgammaf(b[k]);
        float e  = fmaf(a[k], logf(x[k]), fmaf(b[k], logf(1.0f - x[k]), lb));
        float ib = expf(e) * f[k] / a[k];
        r[k] = (x[k] <= 0.0f) ? 0.0f : ((x[k] >= 1.0f) ? 1.0f : ib);
    }

    if (vec) {
        float4 o; o.x = r[0]; o.y = r[1]; o.z = r[2]; o.w = r[3];
        *reinterpret_cast<float4*>(gout + base) = o;
    } else {
        #pragma unroll
        for (int k = 0; k < 4; ++k)
            if (base + k < (long long)n) gout[base + k] = r[k];
    }
}

extern "C" void kernel_launch(void* const* d_in, const int* in_sizes, int n_in,
                              void* d_out, int out_size, void* d_ws, size_t ws_size,
                              hipStream_t stream) {
    (void)n_in; (void)d_ws; (void)ws_size; (void)in_sizes;
    const float* z  = (const float*)d_in[0];
    const float* al = (const float*)d_in[1];
    const float* be = (const float*)d_in[2];
    float* out = (float*)d_out;
    int n  = out_size;
    int n4 = (n + 3) / 4;
    int blocks = (n4 + TPB - 1) / TPB;
    PreciseBetaCDF_kernel<<<blocks, TPB, 0, stream>>>(z, al, be, out, n);
}